// DiscriminatorCell_61598420959602
// MI455X (gfx1250) — compile-verified
//
#include <hip/hip_runtime.h>
#include <hip/hip_bf16.h>

typedef float v2f __attribute__((ext_vector_type(2)));
typedef float v8f __attribute__((ext_vector_type(8)));

#define BATCH 1024
#define SEQL  100
#define FEAT  800
#define ROWS  (BATCH * SEQL)   // 102400 GEMM rows (b,t)

// ---------------------------------------------------------------------------
// Kernel 1: gi[r][n] = |x[r][:]| . w_ih[n][:] + b_ih[n]   (r = b*SEQL + t)
// FP32 WMMA 16x16x4, one wave per 16-row tile, K streamed in chunks of 4,
// grouped 8 chunks (128B per lane-stream) per outer trip with one prefetch.
//
// A-frag layout (16x4 f32): lanes 0-15 hold row M=lane, K={0,1};
//                           lanes 16-31 hold row M=lane-16, K={2,3}.
//   -> per-lane float2 load at xrow + k + (hi?2:0): exact fragment, no selects.
// B-frag layout (4x16 f32): lane n=lane&15 holds col n; lo half K={0,1},
//                           hi half K={2,3} -> contiguous float2 of w_ih row n.
//   Lanes n>=9 feed C columns that are never stored, so they load a clamped
//   (valid but meaningless) row instead of branching -> branch-free inner loop.
// ---------------------------------------------------------------------------
__global__ void __launch_bounds__(256)
gi_gemm_wmma(const float* __restrict__ x,      // (ROWS, FEAT)
             const float* __restrict__ w_ih,   // (9, FEAT)
             const float* __restrict__ b_ih,   // (9)
             float* __restrict__ gi)           // (ROWS, 9)
{
    const int lane = threadIdx.x & 31;
    const int wave = blockIdx.x * (blockDim.x >> 5) + (threadIdx.x >> 5);
    const int row0 = wave << 4;                 // 16 rows per wave
    const int m    = lane & 15;
    const bool hi  = lane >= 16;
    const int koff = hi ? 2 : 0;

    const float* xrow = x + (size_t)(row0 + m) * FEAT + koff;
    const int mclamp  = (m < 9) ? m : 8;        // junk-but-safe row for n>=9
    const float* wrow = w_ih + (size_t)mclamp * FEAT + koff;

    v8f acc = {};
    for (int k = 0; k < FEAT; k += 32) {        // one 128B line per lane-stream
        // stream-ahead hint, once per cacheline (speculative: tail overrun OK)
        __builtin_prefetch(xrow + k + 256, 0, 1);

        #pragma unroll
        for (int kk = 0; kk < 32; kk += 4) {
            const v2f xa = *(const v2f*)(xrow + k + kk);
            v2f a;
            a.x = __builtin_fabsf(xa.x);
            a.y = __builtin_fabsf(xa.y);
            const v2f bf = *(const v2f*)(wrow + k + kk);

            // emits v_wmma_f32_16x16x4_f32
            acc = __builtin_amdgcn_wmma_f32_16x16x4_f32(
                      false, a, false, bf, (short)0, acc, false, false);
        }
    }

    // C/D layout: VGPR v holds M=v (lanes 0-15) / M=8+v (lanes 16-31), N=lane&15
    if (m < 9) {
        const float bias = b_ih[m];
        const int rbase  = row0 + (hi ? 8 : 0);
        #pragma unroll
        for (int v = 0; v < 8; ++v)
            gi[(size_t)(rbase + v) * 9 + m] = acc[v] + bias;
    }
}

// ---------------------------------------------------------------------------
// Kernel 2: per-batch-element GRU scan + conv2 + conv3 + linear + LSTM cell.
// ---------------------------------------------------------------------------
__device__ __forceinline__ float sigmoidf_(float v) {
    return 1.0f / (1.0f + __expf(-v));
}

__global__ void __launch_bounds__(256)
tail_scan(const float* __restrict__ interval,
          const float* __restrict__ gi,          // (BATCH*SEQL, 9), row = b*SEQL+t
          const float* __restrict__ h_prev,      // (BATCH, 10)
          const float* __restrict__ c_prev,      // (BATCH, 10)
          const float* __restrict__ gru_w_hh,    // (9, 3)
          const float* __restrict__ gru_b_hh,    // (9)
          const float* __restrict__ conv2_w,     // (32,1,3)
          const float* __restrict__ conv2_b,     // (32)
          const float* __restrict__ conv3_w,     // (16,32,3)
          const float* __restrict__ conv3_b,     // (16)
          const float* __restrict__ lin_w,       // (10,17)
          const float* __restrict__ lin_b,       // (10)
          const float* __restrict__ lw_ih,       // (40,10)
          const float* __restrict__ lw_hh,       // (40,10)
          const float* __restrict__ lb_ih,       // (40)
          const float* __restrict__ lb_hh,       // (40)
          float* __restrict__ out)               // h_new (BATCH*10) ++ c_new (BATCH*10)
{
    const int b = blockIdx.x * blockDim.x + threadIdx.x;
    if (b >= BATCH) return;

    // ---- GRU recurrence (H=3) ----
    float whh[9][3], bhh[9];
    #pragma unroll
    for (int j = 0; j < 9; ++j) {
        bhh[j] = gru_b_hh[j];
        #pragma unroll
        for (int i = 0; i < 3; ++i) whh[j][i] = gru_w_hh[j * 3 + i];
    }

    float h[3] = {0.0f, 0.0f, 0.0f};
    const float* gr = gi + (size_t)b * SEQL * 9;
    for (int t = 0; t < SEQL; ++t, gr += 9) {
        float gh[9];
        #pragma unroll
        for (int j = 0; j < 9; ++j)
            gh[j] = bhh[j] + whh[j][0]*h[0] + whh[j][1]*h[1] + whh[j][2]*h[2];
        float hn[3];
        #pragma unroll
        for (int n = 0; n < 3; ++n) {
            float r  = sigmoidf_(gr[n]     + gh[n]);
            float z  = sigmoidf_(gr[3 + n] + gh[3 + n]);
            float nn = tanhf(gr[6 + n] + r * gh[6 + n]);
            hn[n] = (1.0f - z) * nn + z * h[n];
        }
        h[0] = hn[0]; h[1] = hn[1]; h[2] = hn[2];
    }

    // ---- conv2 (1->32, k=3, pad=1, relu) fused with conv3 (32->16, k=3) ----
    float z16[16];
    #pragma unroll
    for (int c = 0; c < 16; ++c) z16[c] = conv3_b[c];

    for (int ci = 0; ci < 32; ++ci) {
        const float w0 = conv2_w[ci * 3 + 0];
        const float w1 = conv2_w[ci * 3 + 1];
        const float w2 = conv2_w[ci * 3 + 2];
        const float bb = conv2_b[ci];
        const float y0 = fmaxf(bb + w1*h[0] + w2*h[1], 0.0f);
        const float y1 = fmaxf(bb + w0*h[0] + w1*h[1] + w2*h[2], 0.0f);
        const float y2 = fmaxf(bb + w0*h[1] + w1*h[2], 0.0f);
        #pragma unroll
        for (int c = 0; c < 16; ++c) {
            const float* wp = conv3_w + (size_t)(c * 32 + ci) * 3;
            z16[c] += wp[0]*y0 + wp[1]*y1 + wp[2]*y2;
        }
    }

    // ---- linear 17 -> 10 ----
    float xl[17];
    xl[0] = interval[b];
    #pragma unroll
    for (int c = 0; c < 16; ++c) xl[1 + c] = z16[c];

    float l10[10];
    #pragma unroll
    for (int j = 0; j < 10; ++j) {
        float s = lin_b[j];
        #pragma unroll
        for (int i = 0; i < 17; ++i) s += lin_w[j * 17 + i] * xl[i];
        l10[j] = s;
    }

    // ---- LSTM cell (H=10) ----
    float hp[10], cp[10];
    #pragma unroll
    for (int i = 0; i < 10; ++i) { hp[i] = h_prev[b*10 + i]; cp[i] = c_prev[b*10 + i]; }

    auto gate = [&](int j) {
        float s = lb_ih[j] + lb_hh[j];
        #pragma unroll
        for (int i = 0; i < 10; ++i)
            s += l10[i] * lw_ih[j * 10 + i] + hp[i] * lw_hh[j * 10 + i];
        return s;
    };

    #pragma unroll
    for (int n = 0; n < 10; ++n) {
        const float ig = sigmoidf_(gate(n));
        const float fg = sigmoidf_(gate(10 + n));
        const float gg = tanhf(gate(20 + n));
        const float og = sigmoidf_(gate(30 + n));
        const float cn = fg * cp[n] + ig * gg;
        out[b * 10 + n]              = og * tanhf(cn);   // h_new
        out[BATCH * 10 + b * 10 + n] = cn;               // c_new
    }
}

// ---------------------------------------------------------------------------
extern "C" void kernel_launch(void* const* d_in, const int* in_sizes, int n_in,
                              void* d_out, int out_size, void* d_ws, size_t ws_size,
                              hipStream_t stream)
{
    const float* interval = (const float*)d_in[0];
    const float* sound    = (const float*)d_in[1];
    const float* h_prev   = (const float*)d_in[2];
    const float* c_prev   = (const float*)d_in[3];
    const float* gru_w_ih = (const float*)d_in[4];
    const float* gru_w_hh = (const float*)d_in[5];
    const float* gru_b_ih = (const float*)d_in[6];
    const float* gru_b_hh = (const float*)d_in[7];
    const float* conv2_w  = (const float*)d_in[8];
    const float* conv2_b  = (const float*)d_in[9];
    const float* conv3_w  = (const float*)d_in[10];
    const float* conv3_b  = (const float*)d_in[11];
    const float* lin_w    = (const float*)d_in[12];
    const float* lin_b    = (const float*)d_in[13];
    const float* lw_ih    = (const float*)d_in[14];
    const float* lw_hh    = (const float*)d_in[15];
    const float* lb_ih    = (const float*)d_in[16];
    const float* lb_hh    = (const float*)d_in[17];

    float* gi = (float*)d_ws;   // ROWS * 9 floats = ~3.7 MB of scratch

    // 6400 waves (one per 16-row tile) = 204800 threads = 800 blocks of 256.
    dim3 blk(256);
    dim3 g1((ROWS / 16) * 32 / 256);
    gi_gemm_wmma<<<g1, blk, 0, stream>>>(sound, gru_w_ih, gru_b_ih, gi);

    dim3 g2((BATCH + 255) / 256);
    tail_scan<<<g2, blk, 0, stream>>>(interval, gi, h_prev, c_prev,
                                      gru_w_hh, gru_b_hh,
                                      conv2_w, conv2_b, conv3_w, conv3_b,
                                      lin_w, lin_b,
                                      lw_ih, lw_hh, lb_ih, lb_hh,
                                      (float*)d_out);
}